// UnpadGenAttentionMask_3848290697282
// MI455X (gfx1250) — compile-verified
//
#include <hip/hip_runtime.h>

#define MAX_SEQ  2048
#define HEAD_NUM 16
#define BATCH    8

// ---------------------------------------------------------------------------
// Kernel 1: exclusive prefix of per-batch output sizes (16*s*s), in int64.
// Runs on device because seq_lengths are device data (no host readback
// allowed under graph capture). Trivial work: 8 elements, 1 thread.
// ---------------------------------------------------------------------------
__global__ void compute_offsets_kernel(const int* __restrict__ lens,
                                       long long* __restrict__ offs) {
    if (threadIdx.x == 0) {
        long long acc = 0;
        for (int i = 0; i < BATCH; ++i) {
            offs[i] = acc;
            long long s = (long long)lens[i];
            acc += (long long)HEAD_NUM * s * s;
        }
        offs[BATCH] = acc;
    }
}

// ---------------------------------------------------------------------------
// Kernel 2: one block per (row, batch). Stage the fp16 row into LDS via the
// CDNA5 async global->LDS copy (ASYNCcnt), then each thread computes the
// predicate once and issues 16 head-replicated, lane-coalesced nontemporal
// float stores. Read volume is 1/32 of write volume -> store-BW bound.
// ---------------------------------------------------------------------------
__global__ __launch_bounds__(256)
void unpad_mask_kernel(const _Float16* __restrict__ mask,
                       const int* __restrict__ lens,
                       const long long* __restrict__ offs,
                       float* __restrict__ out) {
    const int b   = blockIdx.y;
    const int row = blockIdx.x;
    const int s   = lens[b];
    if (row >= s) return;

    __shared__ __align__(16) _Float16 srow[MAX_SEQ];

    // --- async copy of the full 2048-element (4096 B) row into LDS ---------
    // 256 threads x 16 B = 4096 B, all 16B-aligned (row base is 4 KiB aligned).
    const _Float16* src = mask + ((size_t)b * MAX_SEQ + (size_t)row) * MAX_SEQ;
    {
        // LDS flat pointers carry the LDS byte offset in their low 32 bits.
        unsigned lds_addr = (unsigned)(size_t)(&srow[0]) + threadIdx.x * 16u;
        unsigned long long gaddr =
            (unsigned long long)(size_t)src + (unsigned long long)threadIdx.x * 16ull;
        asm volatile("global_load_async_to_lds_b128 %0, %1, off"
                     :: "v"(lds_addr), "v"(gaddr)
                     : "memory");
        // Wait for this wave's async copies to land in LDS (ASYNCcnt == 0).
        asm volatile("s_wait_asynccnt 0x0" ::: "memory");
    }
    __syncthreads();   // make every wave's staged bytes visible block-wide

    // --- predicate once, replicate 16x with coalesced NT stores -------------
    const long long ss   = (long long)s * (long long)s;
    const long long base = offs[b] + (long long)row * (long long)s;

    for (int col = (int)threadIdx.x; col < s; col += 256) {
        // fp16 > 0.5 is exact under half->float promotion (0.5 representable).
        float v = ((float)srow[col] > 0.5f) ? 1.0f : 0.0f;
        long long o = base + (long long)col;
#pragma unroll
        for (int h = 0; h < HEAD_NUM; ++h) {
            __builtin_nontemporal_store(v, out + o);  // streaming, bypass L2
            o += ss;
        }
    }
}

// ---------------------------------------------------------------------------
extern "C" void kernel_launch(void* const* d_in, const int* in_sizes, int n_in,
                              void* d_out, int out_size, void* d_ws, size_t ws_size,
                              hipStream_t stream) {
    (void)in_sizes; (void)n_in; (void)out_size; (void)ws_size;

    const _Float16* mask = (const _Float16*)d_in[0];   // fp16 [8,1,2048,2048]
    const int*      lens = (const int*)d_in[1];        // int32 [8]
    long long*      offs = (long long*)d_ws;           // 9 x int64 scratch
    float*          out  = (float*)d_out;              // bool -> float output

    compute_offsets_kernel<<<1, 32, 0, stream>>>(lens, offs);

    dim3 grid(MAX_SEQ, BATCH);
    unpad_mask_kernel<<<grid, 256, 0, stream>>>(mask, lens, offs, out);
}